// NTM_45423574122980
// MI455X (gfx1250) — compile-verified
//
#include <hip/hip_runtime.h>
#include <math.h>
#include <stdint.h>

// ---------------------------------------------------------------------------
// Types for CDNA5 WMMA / async ops
// ---------------------------------------------------------------------------
typedef __attribute__((ext_vector_type(2)))  float    v2f;
typedef __attribute__((ext_vector_type(8)))  float    v8f;
typedef __attribute__((ext_vector_type(4)))  int      v4i;

#define EPS_F 1e-8f

// ---- CDNA5 async global<->LDS support (guarded; fallback = plain copy) ----
#define AS_GLOBAL __attribute__((address_space(1)))
#define AS_LDS    __attribute__((address_space(3)))

#if defined(__has_builtin)
#  if __has_builtin(__builtin_amdgcn_global_load_async_to_lds_b128) && \
      __has_builtin(__builtin_amdgcn_s_wait_asynccnt)
#    define USE_ASYNC_LD 1
#  endif
#  if __has_builtin(__builtin_amdgcn_global_store_async_from_lds_b128) && \
      __has_builtin(__builtin_amdgcn_s_wait_asynccnt)
#    define USE_ASYNC_ST 1
#  endif
#endif
#ifndef USE_ASYNC_LD
#define USE_ASYNC_LD 0
#endif
#ifndef USE_ASYNC_ST
#define USE_ASYNC_ST 0
#endif

__device__ __forceinline__ AS_LDS v4i* to_lds_v4(void* p) {
    // low 32 bits of a flat LDS pointer == wave-relative LDS byte offset
    return (AS_LDS v4i*)(unsigned int)(uintptr_t)p;
}
__device__ __forceinline__ AS_GLOBAL v4i* to_glb_v4(const void* p) {
    return (AS_GLOBAL v4i*)(uintptr_t)p;
}

// ---------------------------------------------------------------------------
// B (K x N, row-major, f32) -> Bt (Np x K, f32, zero-padded rows, transposed)
// ---------------------------------------------------------------------------
__global__ __launch_bounds__(256)
void transpose_padN(const float* __restrict__ src, float* __restrict__ dst,
                    int K, int N, int Np)
{
    const int idx = blockIdx.x * 256 + threadIdx.x;
    if (idx >= Np * K) return;
    const int n = idx / K;
    const int k = idx - n * K;
    const int nc = (n < N) ? n : (N - 1);
    const float v = src[(size_t)k * N + nc];
    dst[idx] = (n < N) ? v : 0.0f;
}

// ---------------------------------------------------------------------------
// Exact f32 WMMA GEMM:  C(MxN) = act(A(MxK) * B^T(NpxK) + bias)
// One wave32 per 16x16 C tile using V_WMMA_F32_16X16X4_F32 (K stepped by 4).
// Fragment layouts per CDNA5 ISA 7.12.2 (32-bit 16x4 A, mirrored for B):
//   A lane (l, half): VGPR0/1 = A[m0+l][k0 + 2*half + 0/1]   (contiguous b64)
//   B lane = column : VGPR0/1 = Bt[n0+l][k0 + 2*half + 0/1]  (contiguous b64)
//   C: vgpr j, lanes 0-15 -> M=j, lanes 16-31 -> M=j+8
// K must be a multiple of 4 (64 and 100 both are).
// ---------------------------------------------------------------------------
__global__ __launch_bounds__(32)
void gemm16_wmma_f32(const float* __restrict__ A,   // M  x K
                     const float* __restrict__ Bt,  // Np x K
                     const float* __restrict__ bias, float* __restrict__ C,
                     int M, int N, int K, int act)
{
    const int lane = threadIdx.x;
    const int half = lane >> 4;          // 0 or 1
    const int l    = lane & 15;
    const int m0   = blockIdx.x << 4;
    const int n0   = blockIdx.y << 4;
    const int col  = n0 + l;

    const float* ap = A  + (size_t)(m0 + l) * K + 2 * half;
    const float* bp = Bt + (size_t)col * K + 2 * half;

    v8f acc = {};
    for (int k0 = 0; k0 < K; k0 += 4) {
        const v2f a = *(const v2f*)(ap + k0);   // 8B aligned (K mult of 4)
        const v2f b = *(const v2f*)(bp + k0);
        acc = __builtin_amdgcn_wmma_f32_16x16x4_f32(
            false, a, false, b, (short)0, acc, false, false);
    }

    if (col < N) {
        const float bv = bias ? bias[col] : 0.0f;
#pragma unroll
        for (int j = 0; j < 8; ++j) {
            const int r = m0 + j + (half ? 8 : 0);
            float v = acc[j] + bv;
            if (act == 1) v = 1.0f / (1.0f + expf(-v));
            C[(size_t)r * N + col] = v;
        }
    }
}

// ---------------------------------------------------------------------------
// Fused NTM step: one workgroup (512 threads = 16 waves) per batch row.
// memory[b] tile (512x64 f32) lives in LDS, row stride 68 floats (16B-aligned
// rows -> b128 LDS/async ops; 2-way bank conflict on row dots, none on the
// column reduction).
// ---------------------------------------------------------------------------
#define NROW 512
#define MD   64
#define LSTR 68
#define NTH  512

__device__ __forceinline__ float softplusf(float x) {
    return (x > 20.0f) ? x : log1pf(expf(x));
}
__device__ __forceinline__ float sigmoidf(float x) {
    return 1.0f / (1.0f + expf(-x));
}

__device__ __forceinline__ float wave_sum(float v) {
#pragma unroll
    for (int off = 16; off > 0; off >>= 1) v += __shfl_xor(v, off, 32);
    return v;
}
__device__ __forceinline__ float wave_max(float v) {
#pragma unroll
    for (int off = 16; off > 0; off >>= 1) v = fmaxf(v, __shfl_xor(v, off, 32));
    return v;
}

__device__ __forceinline__ float block_sum(float* red, int tid, float v) {
    v = wave_sum(v);
    if ((tid & 31) == 0) red[tid >> 5] = v;
    __syncthreads();
    if (tid < 32) {
        float t = (tid < 16) ? red[tid] : 0.0f;
        t = wave_sum(t);
        if (tid == 0) red[0] = t;
    }
    __syncthreads();
    const float r = red[0];
    __syncthreads();
    return r;
}
__device__ __forceinline__ float block_max(float* red, int tid, float v) {
    v = wave_max(v);
    if ((tid & 31) == 0) red[tid >> 5] = v;
    __syncthreads();
    if (tid < 32) {
        float t = (tid < 16) ? red[tid] : -INFINITY;
        t = wave_max(t);
        if (tid == 0) red[0] = t;
    }
    __syncthreads();
    const float r = red[0];
    __syncthreads();
    return r;
}

// NTM addressing: cosine sim -> softmax -> gate -> circular 3-tap shift ->
// sharpen (pow gamma) -> renormalize.  Executed by all 512 threads, n = tid.
__device__ float ntm_address(float dot, float nrm, float kn, float beta,
                             float gate, float s0, float s1, float s2,
                             float gamma, const float* __restrict__ prev,
                             float* wg, float* red, int n)
{
    const float mn = fmaxf(sqrtf(nrm), EPS_F);
    const float Kc = dot / (kn * mn);
    const float s  = beta * Kc;
    const float mx = block_max(red, n, s);
    const float e  = expf(s - mx);
    const float se = block_sum(red, n, e);
    const float wc = e / se;
    const float wgl = gate * wc + (1.0f - gate) * prev[n];
    wg[n] = wgl;
    __syncthreads();
    const float tw = s2 * wg[(n + NROW - 1) & (NROW - 1)] + s1 * wgl +
                     s0 * wg[(n + 1) & (NROW - 1)];
    __syncthreads();   // wg free for reuse after this point
    const float twg = powf(fmaxf(tw, 0.0f), gamma);
    const float ts  = block_sum(red, n, twg);
    return twg / (EPS_F + ts);
}

__global__ __launch_bounds__(NTH, 2)
void ntm_fused(const float* __restrict__ memory,
               const float* __restrict__ read_attn,
               const float* __restrict__ write_attn,
               const float* __restrict__ oW, const float* __restrict__ ob,
               const float* __restrict__ wp,   // (B,70)  write head params
               const float* __restrict__ wep,  // (B,128) erase/add params
               const float* __restrict__ rp,   // (B,70)  read head params
               float* __restrict__ o_out, float* __restrict__ o_mem2,
               float* __restrict__ o_wr, float* __restrict__ o_ww)
{
    extern __shared__ float smem[];
    float* memT = smem;                    // NROW * LSTR
    float* kw   = memT + NROW * LSTR;      // 64
    float* kr   = kw + MD;                 // 64
    float* ev   = kr + MD;                 // 64
    float* av   = ev + MD;                 // 64
    float* wg   = av + MD;                 // 512
    float* wcur = wg + NROW;               // 512
    float* red  = wcur + NROW;             // 512
    float* rvec = red + NROW;              // 64
    float* scal = rvec + MD;               // 16

    const int tid = threadIdx.x;
    const int n   = tid;
    const int b   = blockIdx.x;
    const size_t mbase = (size_t)b * NROW * MD;

    // ---- tile load: CDNA5 async global->LDS (ASYNCcnt) where available ----
#if USE_ASYNC_LD
    for (int i = tid; i < (NROW * MD) / 4; i += NTH) {
        const int f = i << 2;
        const int row = f >> 6;
        const int col = f & 63;
        __builtin_amdgcn_global_load_async_to_lds_b128(
            to_glb_v4(memory + mbase + f),
            to_lds_v4(memT + row * LSTR + col), 0, 0);
    }
    __builtin_amdgcn_s_wait_asynccnt(0);
#else
    for (int i = tid; i < (NROW * MD) / 4; i += NTH) {
        const int f = i << 2;
        const int row = f >> 6;
        const int col = f & 63;
        *(float4*)(memT + row * LSTR + col) =
            *(const float4*)(memory + mbase + f);
    }
#endif

    // ---- per-batch head parameters ---------------------------------------
    if (tid < MD) {
        kw[tid] = wp[b * 70 + tid];
        kr[tid] = rp[b * 70 + tid];
        ev[tid] = sigmoidf(wep[b * 128 + tid]);
        av[tid] = wep[b * 128 + MD + tid];
    }
    if (tid == 0) {   // write-head scalars
        const float* p = wp + b * 70;
        scal[1] = softplusf(p[64]);
        scal[2] = sigmoidf(p[65]);
        const float a0 = p[66], a1 = p[67], a2 = p[68];
        const float mx = fmaxf(a0, fmaxf(a1, a2));
        const float e0 = expf(a0 - mx), e1 = expf(a1 - mx), e2 = expf(a2 - mx);
        const float se = e0 + e1 + e2;
        scal[3] = e0 / se; scal[4] = e1 / se; scal[5] = e2 / se;
        scal[6] = 1.0f + softplusf(p[69]);
    }
    if (tid == 1) {   // read-head scalars
        const float* p = rp + b * 70;
        scal[8]  = softplusf(p[64]);
        scal[9]  = sigmoidf(p[65]);
        const float a0 = p[66], a1 = p[67], a2 = p[68];
        const float mx = fmaxf(a0, fmaxf(a1, a2));
        const float e0 = expf(a0 - mx), e1 = expf(a1 - mx), e2 = expf(a2 - mx);
        const float se = e0 + e1 + e2;
        scal[10] = e0 / se; scal[11] = e1 / se; scal[12] = e2 / se;
        scal[13] = 1.0f + softplusf(p[69]);
    }
    __syncthreads();
    if (tid == 0) {
        float s = 0.0f;
        for (int m = 0; m < MD; ++m) s += kw[m] * kw[m];
        scal[0] = fmaxf(sqrtf(s), EPS_F);
    }
    if (tid == 1) {
        float s = 0.0f;
        for (int m = 0; m < MD; ++m) s += kr[m] * kr[m];
        scal[7] = fmaxf(sqrtf(s), EPS_F);
    }
    __syncthreads();

    // ---- write addressing (pre-update memory) ----------------------------
    float* mrow = memT + n * LSTR;
    float dot = 0.0f, nrm = 0.0f;
#pragma unroll 8
    for (int m = 0; m < MD; ++m) {
        const float v = mrow[m];
        dot += v * kw[m];
        nrm += v * v;
    }
    const float ww = ntm_address(dot, nrm, scal[0], scal[1], scal[2],
                                 scal[3], scal[4], scal[5], scal[6],
                                 write_attn + (size_t)b * NROW, wg, red, n);
    o_ww[(size_t)b * NROW + n] = ww;

    // ---- memory update; fused read-head dot/norm -------------------------
    float dotr = 0.0f, nrmr = 0.0f;
#pragma unroll 8
    for (int m = 0; m < MD; ++m) {
        float v = mrow[m];
        v = v * (1.0f - ww * ev[m]) + ww * av[m];
        mrow[m] = v;
        dotr += v * kr[m];
        nrmr += v * v;
    }
    __syncthreads();

    // ---- mem2 store: async LDS->global where available -------------------
#if USE_ASYNC_ST
    for (int i = tid; i < (NROW * MD) / 4; i += NTH) {
        const int f = i << 2;
        const int row = f >> 6;
        const int col = f & 63;
        __builtin_amdgcn_global_store_async_from_lds_b128(
            to_glb_v4(o_mem2 + mbase + f),
            to_lds_v4(memT + row * LSTR + col), 0, 0);
    }
#else
    for (int i = tid; i < (NROW * MD) / 4; i += NTH) {
        const int f = i << 2;
        const int row = f >> 6;
        const int col = f & 63;
        *(float4*)(o_mem2 + mbase + f) =
            *(const float4*)(memT + row * LSTR + col);
    }
#endif

    // ---- read addressing (updated memory, still in LDS) ------------------
    const float wr = ntm_address(dotr, nrmr, scal[7], scal[8], scal[9],
                                 scal[10], scal[11], scal[12], scal[13],
                                 read_attn + (size_t)b * NROW, wg, red, n);
    wcur[n] = wr;
    o_wr[(size_t)b * NROW + n] = wr;
    __syncthreads();

    // ---- r[m] = sum_n wr[n] * mem2[n][m] ---------------------------------
    {
        const int m = tid & 63;
        const int g = tid >> 6;
        float acc = 0.0f;
#pragma unroll 8
        for (int j = 0; j < 64; ++j) {
            const int nn = (g << 6) + j;
            acc += wcur[nn] * memT[nn * LSTR + m];
        }
        red[tid] = acc;
    }
    __syncthreads();
    if (tid < MD) {
        float s = 0.0f;
#pragma unroll
        for (int g = 0; g < 8; ++g) s += red[(g << 6) + tid];
        rvec[tid] = s;
    }
    __syncthreads();

    // ---- out = sigmoid(r @ oW + ob) --------------------------------------
    if (tid < MD) {
        float s = ob[tid];
#pragma unroll 8
        for (int m = 0; m < MD; ++m) s += rvec[m] * oW[m * MD + tid];
        o_out[(size_t)b * MD + tid] = sigmoidf(s);
    }

#if USE_ASYNC_ST
    __builtin_amdgcn_s_wait_asynccnt(0);
#endif
}

// ---------------------------------------------------------------------------
// Host-side launch
// ---------------------------------------------------------------------------
extern "C" void kernel_launch(void* const* d_in, const int* in_sizes, int n_in,
                              void* d_out, int out_size, void* d_ws, size_t ws_size,
                              hipStream_t stream)
{
    const float* x      = (const float*)d_in[0];
    const float* memory = (const float*)d_in[1];
    const float* rattn  = (const float*)d_in[2];
    const float* wattn  = (const float*)d_in[3];
    const float* cW     = (const float*)d_in[4];
    const float* cb     = (const float*)d_in[5];
    const float* oW     = (const float*)d_in[6];
    const float* ob     = (const float*)d_in[7];
    const float* whW    = (const float*)d_in[8];
    const float* whb    = (const float*)d_in[9];
    const float* wheW   = (const float*)d_in[10];
    const float* wheb   = (const float*)d_in[11];
    const float* rhW    = (const float*)d_in[12];
    const float* rhb    = (const float*)d_in[13];

    const int IN = 64, H = 100, MDIM = 64, NR = 512;
    const int B = in_sizes[0] / IN;           // 2048
    const int Np_H = 112, Np_70 = 80, Np_128 = 128;   // N padded to mult of 16

    // workspace layout (all f32)
    char* w = (char*)d_ws;
    float* h     = (float*)w;  w += (size_t)B * H   * 4;
    float* wp    = (float*)w;  w += (size_t)B * 70  * 4;
    float* wep   = (float*)w;  w += (size_t)B * 128 * 4;
    float* rp    = (float*)w;  w += (size_t)B * 70  * 4;
    float* cWt   = (float*)w;  w += (size_t)Np_H   * IN * 4;
    float* whWt  = (float*)w;  w += (size_t)Np_70  * H  * 4;
    float* wheWt = (float*)w;  w += (size_t)Np_128 * H  * 4;
    float* rhWt  = (float*)w;  w += (size_t)Np_70  * H  * 4;

    auto cdiv = [](int a, int b) { return (a + b - 1) / b; };

    // weight transposes (f32, N-padded) for contiguous B fragments
    transpose_padN<<<cdiv(Np_H * IN, 256),  256, 0, stream>>>(cW,   cWt,   IN, H,   Np_H);
    transpose_padN<<<cdiv(Np_70 * H, 256),  256, 0, stream>>>(whW,  whWt,  H,  70,  Np_70);
    transpose_padN<<<cdiv(Np_128 * H, 256), 256, 0, stream>>>(wheW, wheWt, H,  128, Np_128);
    transpose_padN<<<cdiv(Np_70 * H, 256),  256, 0, stream>>>(rhW,  rhWt,  H,  70,  Np_70);

    const dim3 wmma_blk(32);
    gemm16_wmma_f32<<<dim3(B / 16, Np_H / 16), wmma_blk, 0, stream>>>(
        x, cWt, cb, h, B, H, IN, 0);
    gemm16_wmma_f32<<<dim3(B / 16, Np_70 / 16), wmma_blk, 0, stream>>>(
        h, whWt, whb, wp, B, 70, H, 0);
    gemm16_wmma_f32<<<dim3(B / 16, Np_128 / 16), wmma_blk, 0, stream>>>(
        h, wheWt, wheb, wep, B, 128, H, 0);
    gemm16_wmma_f32<<<dim3(B / 16, Np_70 / 16), wmma_blk, 0, stream>>>(
        h, rhWt, rhb, rp, B, 70, H, 0);

    // outputs, flattened in reference return order: out | mem2 | w_r | w_w
    float* o_out  = (float*)d_out;
    float* o_mem2 = o_out  + (size_t)B * MDIM;
    float* o_wr   = o_mem2 + (size_t)B * NR * MDIM;
    float* o_ww   = o_wr   + (size_t)B * NR;

    const size_t smem_bytes =
        (size_t)(NR * LSTR + 4 * MDIM + 3 * NR + MDIM + 16) * sizeof(float);

    ntm_fused<<<dim3(B), dim3(NTH), smem_bytes, stream>>>(
        memory, rattn, wattn, oW, ob, wp, wep, rp,
        o_out, o_mem2, o_wr, o_ww);
}